// RiemannianTangentBridge_64759516889557
// MI455X (gfx1250) — compile-verified
//
#include <hip/hip_runtime.h>

// Problem constants (from reference): B=2048, D=256, K=32
#define BB 2048
#define DD 256
#define KK 32
#define JITTER 1e-6f

typedef __attribute__((ext_vector_type(2))) float v2f;
typedef __attribute__((ext_vector_type(8))) float v8f;

__device__ __forceinline__ v8f wmma4(v2f a, v2f b, v8f c) {
  // V_WMMA_F32_16X16X4_F32: D(16x16 f32) = A(16x4 f32) x B(4x16 f32) + C
  return __builtin_amdgcn_wmma_f32_16x16x4_f32(
      /*neg_a=*/false, a, /*neg_b=*/false, b,
      /*c_mod=*/(short)0, c, /*reuse_a=*/false, /*reuse_b=*/false);
}

// ---------------------------------------------------------------------------
// Kernel 0: M = qr(M_raw).Q  via modified Gram-Schmidt. One block, 256 threads
// (one thread per row of the 256x32 matrix; row kept in registers).
// ---------------------------------------------------------------------------
__global__ __launch_bounds__(256) void qr_kernel(const float* __restrict__ Mraw,
                                                 float* __restrict__ Mout) {
  __shared__ float red[8];
  const int t = threadIdx.x;          // row index 0..255
  const int lane = t & 31, wave = t >> 5;
  float a[KK];
#pragma unroll
  for (int j = 0; j < KK; ++j) a[j] = Mraw[t * KK + j];

  for (int j = 0; j < KK; ++j) {
    for (int i = 0; i < j; ++i) {
      float p = a[i] * a[j];
      for (int off = 16; off; off >>= 1) p += __shfl_xor(p, off);
      __syncthreads();                 // prior reads of red[] done
      if (lane == 0) red[wave] = p;
      __syncthreads();
      float r = 0.f;
#pragma unroll
      for (int w = 0; w < 8; ++w) r += red[w];
      a[j] -= r * a[i];
    }
    float p = a[j] * a[j];
    for (int off = 16; off; off >>= 1) p += __shfl_xor(p, off);
    __syncthreads();
    if (lane == 0) red[wave] = p;
    __syncthreads();
    float nrm = 0.f;
#pragma unroll
    for (int w = 0; w < 8; ++w) nrm += red[w];
    a[j] *= rsqrtf(fmaxf(nrm, 1e-30f));
  }
#pragma unroll
  for (int j = 0; j < KK; ++j) Mout[t * KK + j] = a[j];
}

// ---------------------------------------------------------------------------
// Kernel 1: full per-batch Grassmannian log map. One block (128 thr = 4 waves)
// per batch element. Everything staged in LDS; W read once, Delta written once.
// ---------------------------------------------------------------------------
struct alignas(16) SMEM {
  float W[DD * KK];    // 32 KB  W_b
  float ME[DD * KK];   // 32 KB  M, overwritten in-place by E
  float GJ[KK * 64];   // 8 KB   [G+eps*I | I] -> [I | G^-1]
  float C[KK * KK];    // 4 KB   E^T E, diagonalized in place
  float V[KK * KK];    // 4 KB   eigenvectors
  float T[KK * KK];    // 4 KB   scratch for Jacobi / V*diag(f)
  float F[KK * KK];    // 4 KB   V f(L) V^T
  float rowp[64];
  float colp[KK];
  float alpha[KK], beta_[KK], fdiag[KK];
  int   partner[KK];
  int   piv;
  float pivinv;
};

__global__ __launch_bounds__(128) void grass_kernel(const float* __restrict__ Wg,
                                                    const float* __restrict__ Mg,
                                                    float* __restrict__ Out) {
  __shared__ SMEM s;
  const int tid = threadIdx.x;
  const int lane = tid & 31, wv = tid >> 5;
  const int lh = lane >> 4;            // half-wave select (0/1)
  const int lm = lane & 15;            // position within half
  const int bidx = blockIdx.x;
  const float* __restrict__ Wb = Wg + (size_t)bidx * (DD * KK);

  // ---- stage W_b and M into LDS (float4) ----
  {
    const float4* w4 = (const float4*)Wb;
    const float4* m4 = (const float4*)Mg;
    float4* sw4 = (float4*)s.W;
    float4* sm4 = (float4*)s.ME;
    for (int i = tid; i < DD * KK / 4; i += 128) { sw4[i] = w4[i]; sm4[i] = m4[i]; }
  }
  __syncthreads();

  // ---- G = M^T W (32x32, reduce over D=256); 4 waves x one 16x16 tile ----
  {
    const int i0 = (wv >> 1) * 16, j0 = (wv & 1) * 16;
    v8f acc = {};
    for (int d0 = 0; d0 < DD; d0 += 4) {
      const int r0 = d0 + 2 * lh;      // A: k = v + 2*lh, row = lm
      v2f av, bv;
      av.x = s.ME[(r0 + 0) * KK + i0 + lm];
      av.y = s.ME[(r0 + 1) * KK + i0 + lm];
      bv.x = s.W[(r0 + 0) * KK + j0 + lm];
      bv.y = s.W[(r0 + 1) * KK + j0 + lm];
      acc = wmma4(av, bv, acc);
    }
#pragma unroll
    for (int v = 0; v < 8; ++v) {      // D: row = v + 8*lh, col = lm
      const int r = i0 + v + 8 * lh, c = j0 + lm;
      s.GJ[r * 64 + c] = acc[v] + ((r == c) ? JITTER : 0.0f);
    }
  }
  __syncthreads();
  for (int i = tid; i < KK * KK; i += 128) {  // augment with identity
    const int r = i >> 5, c = i & 31;
    s.GJ[r * 64 + 32 + c] = (r == c) ? 1.0f : 0.0f;
  }
  __syncthreads();

  // ---- Gauss-Jordan with partial pivoting: GJ -> [I | G^-1] ----
  for (int p = 0; p < KK; ++p) {
    if (tid == 0) {
      int best = p; float bv = fabsf(s.GJ[p * 64 + p]);
      for (int r = p + 1; r < KK; ++r) {
        const float v = fabsf(s.GJ[r * 64 + p]);
        if (v > bv) { bv = v; best = r; }
      }
      s.piv = best;
    }
    __syncthreads();
    const int pr = s.piv;
    if (pr != p && tid < 64) {
      const float tv = s.GJ[p * 64 + tid];
      s.GJ[p * 64 + tid] = s.GJ[pr * 64 + tid];
      s.GJ[pr * 64 + tid] = tv;
    }
    __syncthreads();
    if (tid == 0) s.pivinv = 1.0f / s.GJ[p * 64 + p];
    __syncthreads();
    if (tid < 64) s.rowp[tid] = s.GJ[p * 64 + tid] * s.pivinv;
    if (tid >= 64 && tid < 96) s.colp[tid - 64] = s.GJ[(tid - 64) * 64 + p];
    __syncthreads();
    if (tid < 64) s.GJ[p * 64 + tid] = s.rowp[tid];
    for (int e = tid; e < KK * 64; e += 128) {
      const int r = e >> 6, c = e & 63;
      if (r != p) s.GJ[e] -= s.colp[r] * s.rowp[c];
    }
    __syncthreads();
  }

  // ---- E = W * G^-1 - M, written over M in place (256x32); 32 tiles ----
  for (int t8 = 0; t8 < 8; ++t8) {
    const int t = wv + t8 * 4;
    const int i0 = (t >> 1) * 16, j0 = (t & 1) * 16;
    v8f acc = {};
    for (int k0 = 0; k0 < KK; k0 += 4) {
      const int kk = k0 + 2 * lh;
      v2f av, bv;
      av.x = s.W[(i0 + lm) * KK + kk + 0];
      av.y = s.W[(i0 + lm) * KK + kk + 1];
      bv.x = s.GJ[(kk + 0) * 64 + 32 + j0 + lm];
      bv.y = s.GJ[(kk + 1) * 64 + 32 + j0 + lm];
      acc = wmma4(av, bv, acc);
    }
#pragma unroll
    for (int v = 0; v < 8; ++v) {
      const int r = i0 + v + 8 * lh, c = j0 + lm;
      s.ME[r * KK + c] = acc[v] - s.ME[r * KK + c];   // E = Y - M (in place)
    }
  }
  __syncthreads();

  // ---- C = E^T E (32x32) ----
  {
    const int i0 = (wv >> 1) * 16, j0 = (wv & 1) * 16;
    v8f acc = {};
    for (int d0 = 0; d0 < DD; d0 += 4) {
      const int r0 = d0 + 2 * lh;
      v2f av, bv;
      av.x = s.ME[(r0 + 0) * KK + i0 + lm];
      av.y = s.ME[(r0 + 1) * KK + i0 + lm];
      bv.x = s.ME[(r0 + 0) * KK + j0 + lm];
      bv.y = s.ME[(r0 + 1) * KK + j0 + lm];
      acc = wmma4(av, bv, acc);
    }
#pragma unroll
    for (int v = 0; v < 8; ++v)
      s.C[(i0 + v + 8 * lh) * KK + j0 + lm] = acc[v];
  }
  for (int i = tid; i < KK * KK; i += 128)
    s.V[i] = ((i >> 5) == (i & 31)) ? 1.0f : 0.0f;
  __syncthreads();

  // ---- Jacobi eigensolver on C (parallel round-robin: 16 pairs/round) ----
  for (int sweep = 0; sweep < 7; ++sweep) {
    for (int rr = 0; rr < 31; ++rr) {
      if (tid < 16) {
        int p, q;
        if (tid == 0) { p = 31; q = rr; }
        else { p = (rr + tid) % 31; q = (rr + 31 - tid) % 31; }
        const float app = s.C[p * KK + p];
        const float aqq = s.C[q * KK + q];
        const float apq = s.C[p * KK + q];
        float c = 1.0f, sn = 0.0f;
        if (fabsf(apq) > 1e-12f) {
          const float tau = (aqq - app) / (2.0f * apq);
          const float tt = (tau >= 0.0f ? 1.0f : -1.0f) /
                           (fabsf(tau) + sqrtf(1.0f + tau * tau));
          c = rsqrtf(1.0f + tt * tt);
          sn = tt * c;
        }
        s.partner[p] = q; s.partner[q] = p;
        s.alpha[p] = c;  s.alpha[q] = c;
        s.beta_[p] = -sn; s.beta_[q] = sn;
      }
      __syncthreads();
      for (int e = tid; e < KK * KK; e += 128) {      // T = J^T C  (rows)
        const int i = e >> 5, j = e & 31;
        s.T[e] = s.alpha[i] * s.C[e] + s.beta_[i] * s.C[s.partner[i] * KK + j];
      }
      __syncthreads();
      for (int e = tid; e < KK * KK; e += 128) {      // C = T J    (cols)
        const int i = e >> 5, j = e & 31;
        s.C[e] = s.alpha[j] * s.T[e] + s.beta_[j] * s.T[i * KK + s.partner[j]];
      }
      __syncthreads();
      for (int e = tid; e < KK * KK; e += 128) s.T[e] = s.V[e];
      __syncthreads();
      for (int e = tid; e < KK * KK; e += 128) {      // V = V J
        const int i = e >> 5, j = e & 31;
        s.V[e] = s.alpha[j] * s.T[e] + s.beta_[j] * s.T[i * KK + s.partner[j]];
      }
      __syncthreads();
    }
  }

  // ---- f(lambda) = atan(sqrt(l))/sqrt(l);  T = V*diag(f) ----
  if (tid < KK) {
    const float lam = fmaxf(s.C[tid * KK + tid], 0.0f);
    const float sv = sqrtf(lam);
    s.fdiag[tid] = (sv > 1e-6f) ? (atanf(sv) / sv) : (1.0f - lam * (1.0f / 3.0f));
  }
  __syncthreads();
  for (int e = tid; e < KK * KK; e += 128) s.T[e] = s.V[e] * s.fdiag[e & 31];
  __syncthreads();

  // ---- F = (V diag(f)) * V^T ----
  {
    const int i0 = (wv >> 1) * 16, j0 = (wv & 1) * 16;
    v8f acc = {};
    for (int k0 = 0; k0 < KK; k0 += 4) {
      const int kk = k0 + 2 * lh;
      v2f av, bv;
      av.x = s.T[(i0 + lm) * KK + kk + 0];
      av.y = s.T[(i0 + lm) * KK + kk + 1];
      bv.x = s.V[(j0 + lm) * KK + kk + 0];   // B = V^T
      bv.y = s.V[(j0 + lm) * KK + kk + 1];
      acc = wmma4(av, bv, acc);
    }
#pragma unroll
    for (int v = 0; v < 8; ++v)
      s.F[(i0 + v + 8 * lh) * KK + j0 + lm] = acc[v];
  }
  __syncthreads();

  // ---- Delta = E * F, streamed straight to global output ----
  float* __restrict__ Ob = Out + (size_t)bidx * (DD * KK);
  for (int t8 = 0; t8 < 8; ++t8) {
    const int t = wv + t8 * 4;
    const int i0 = (t >> 1) * 16, j0 = (t & 1) * 16;
    v8f acc = {};
    for (int k0 = 0; k0 < KK; k0 += 4) {
      const int kk = k0 + 2 * lh;
      v2f av, bv;
      av.x = s.ME[(i0 + lm) * KK + kk + 0];
      av.y = s.ME[(i0 + lm) * KK + kk + 1];
      bv.x = s.F[(kk + 0) * KK + j0 + lm];
      bv.y = s.F[(kk + 1) * KK + j0 + lm];
      acc = wmma4(av, bv, acc);
    }
#pragma unroll
    for (int v = 0; v < 8; ++v)
      Ob[(i0 + v + 8 * lh) * KK + j0 + lm] = acc[v];
  }
}

// ---------------------------------------------------------------------------
extern "C" void kernel_launch(void* const* d_in, const int* in_sizes, int n_in,
                              void* d_out, int out_size, void* d_ws, size_t ws_size,
                              hipStream_t stream) {
  (void)in_sizes; (void)n_in; (void)out_size; (void)ws_size;
  const float* W    = (const float*)d_in[0];   // (B, D, K) f32
  const float* Mraw = (const float*)d_in[1];   // (D, K)    f32
  float* Mq = (float*)d_ws;                    // orthonormal M (D*K floats)

  qr_kernel<<<1, 256, 0, stream>>>(Mraw, Mq);
  grass_kernel<<<BB, 128, 0, stream>>>(W, Mq, (float*)d_out);
}